// Model_24180665876778
// MI455X (gfx1250) — compile-verified
//
#include <hip/hip_runtime.h>
#include <hip/hip_bf16.h>
#include <math.h>

// ---------------- model constants ----------------
#define Bc    16
#define Lc    512
#define Nc    21
#define Dc    128
#define Hc    8
#define Mc    32      // kept fourier modes
#define DFFc  512
#define PREDc 96
#define NLc   2
#define Ec    16      // D/H
#define BNc   (Bc*Nc)         // 336
#define ROWS  (BNc*Lc)        // 172032 token rows
#define FR    (BNc*Dc)        // 43008 frequency rows (bn,h,e)

#define KB    32              // GEMM K-block staged through LDS
#define LDP   34              // padded LDS row stride (even, bank-conflict-free)

typedef float v2f __attribute__((ext_vector_type(2)));
typedef float v4f __attribute__((ext_vector_type(4)));
typedef float v8f __attribute__((ext_vector_type(8)));

static __device__ __forceinline__ v8f wmma4(v2f a, v2f b, v8f c) {
  // V_WMMA_F32_16X16X4_F32 : D(16x16 f32) = A(16x4 f32) * B(4x16 f32) + C
  return __builtin_amdgcn_wmma_f32_16x16x4_f32(false, a, false, b, (short)0, c, false, false);
}

static __device__ __forceinline__ float gelu_exact(float x) {
  return 0.5f * x * (1.0f + erff(x * 0.70710678118654752f));
}

// ---------------- series decomposition (moving avg K=25, edge pad) ----------
__global__ void decomp_k(const float* __restrict__ in, float* __restrict__ seas,
                         float* __restrict__ trend, int O, int C) {
  int idx = blockIdx.x * blockDim.x + threadIdx.x;
  if (idx >= O * C) return;
  int c = idx % C;
  int o = idx / C;
  const float* base = in + (size_t)o * Lc * C + c;
  float* sb = seas + (size_t)o * Lc * C + c;
  float* tb = trend ? (trend + (size_t)o * Lc * C + c) : nullptr;
  float first = base[0];
  float sum = 12.0f * first;            // 12 replicated left-pad samples
  for (int j = 0; j < 13; ++j) sum += base[(size_t)j * C];
  const float inv = 1.0f / 25.0f;
  for (int l = 0; l < Lc; ++l) {
    float mean = sum * inv;
    float xv = base[(size_t)l * C];
    sb[(size_t)l * C] = xv - mean;
    if (tb) tb[(size_t)l * C] = mean;
    int ja = (l + 13 > Lc - 1) ? (Lc - 1) : (l + 13);
    int jr = (l - 12 < 0) ? 0 : (l - 12);
    sum += base[(size_t)ja * C] - base[(size_t)jr * C];
  }
}

// ---------------- embedding: tokenconv(k=3, circular) + time linear --------
__global__ void embed_k(const float* __restrict__ seas0, const float* __restrict__ xmark,
                        const float* __restrict__ tokw, const float* __restrict__ timew,
                        float* __restrict__ X) {
  int idx = blockIdx.x * blockDim.x + threadIdx.x;
  if (idx >= BNc * Lc * Dc) return;
  int d  = idx & (Dc - 1);
  int l  = (idx >> 7) & (Lc - 1);
  int bn = idx >> 16;
  int b = bn / Nc;
  int n = bn - b * Nc;
  int lm1 = (l + Lc - 1) & (Lc - 1);
  int lp1 = (l + 1) & (Lc - 1);
  const float* sp = seas0 + (size_t)b * Lc * Nc + n;
  float s_m = sp[(size_t)lm1 * Nc];
  float s_c = sp[(size_t)l   * Nc];
  float s_p = sp[(size_t)lp1 * Nc];
  float w0 = tokw[d * 3 + 0], w1 = tokw[d * 3 + 1], w2 = tokw[d * 3 + 2];
  const float* mk = xmark + ((size_t)b * Lc + l) * 4;
  const float* tw = timew + (size_t)d * 4;
  float tv = mk[0] * tw[0] + mk[1] * tw[1] + mk[2] * tw[2] + mk[3] * tw[3];
  X[idx] = s_m * w0 + s_c * w1 + s_p * w2 + tv;
}

// ---------------- tiled (O,R,C)->(O,C,R) transpose --------------------------
__global__ void tr021_k(const float* __restrict__ in, float* __restrict__ out, int R, int C) {
  __shared__ float t[32][33];
  int o  = blockIdx.z;
  int c0 = blockIdx.x * 32;
  int r0 = blockIdx.y * 32;
  const float* ib = in + (size_t)o * R * C;
  float* ob = out + (size_t)o * R * C;
  for (int i = threadIdx.y; i < 32; i += 8)
    t[i][threadIdx.x] = ib[(size_t)(r0 + i) * C + c0 + threadIdx.x];
  __syncthreads();
  for (int i = threadIdx.y; i < 32; i += 8)
    ob[(size_t)(c0 + i) * R + r0 + threadIdx.x] = t[threadIdx.x][i];
}

// ---------------- DFT / iDFT basis generation -------------------------------
__global__ void basis_k(float* __restrict__ W1, float* __restrict__ W3) {
  int idx = blockIdx.x * blockDim.x + threadIdx.x;
  const float w0 = 6.2831853071795864769f / (float)Lc;
  if (idx < 64 * Lc) {                       // W1: (64 x 512), rows [cos(0..31) | -sin(0..31)]
    int row = idx >> 9;
    int l = idx & (Lc - 1);
    int m = row & 31;
    int tt = (m * l) & (Lc - 1);
    float a = (float)tt * w0;
    W1[idx] = (row < 32) ? cosf(a) : -sinf(a);
  } else if (idx < 64 * Lc + Lc * 64) {      // W3: (512 x 64), irfft weights
    int j = idx - 64 * Lc;
    int l = j >> 6;
    int c = j & 63;
    int m = c & 31;
    int tt = (m * l) & (Lc - 1);
    float a = (float)tt * w0;
    float v;
    if (c < 32) v = ((m == 0) ? 1.0f : 2.0f) * cosf(a) * (1.0f / (float)Lc);
    else        v = (m == 0) ? 0.0f : (-2.0f / (float)Lc) * sinf(a);
    W3[j] = v;
  }
}

// ---------------- complex per-mode mixing over e ----------------------------
// S1 rows: (bn*D + h*E + e), cols [Re(0..31) | Im(0..31)] -> S2 rows (bn*D + h*E + o)
__global__ void mix_k(const float* __restrict__ S1, const float* __restrict__ wr,
                      const float* __restrict__ wi, float* __restrict__ S2) {
  int idx = blockIdx.x * blockDim.x + threadIdx.x;
  if (idx >= BNc * Hc * Ec * Mc) return;
  int m  = idx & 31;
  int o  = (idx >> 5) & 15;
  int h  = (idx >> 9) & 7;
  int bn = idx >> 12;
  const float* xb  = S1 + ((size_t)(bn * Dc + h * Ec)) * 64 + m;          // e = 0
  const float* wrb = wr + ((size_t)(h * Ec) * Ec + o) * Mc + m;           // e = 0
  const float* wib = wi + ((size_t)(h * Ec) * Ec + o) * Mc + m;
  float ar = 0.0f, ai = 0.0f;
  #pragma unroll
  for (int e = 0; e < Ec; ++e) {
    float xr = xb[(size_t)e * 64];
    float xi = xb[(size_t)e * 64 + 32];
    float wrv = wrb[(size_t)e * (Ec * Mc)];
    float wiv = wib[(size_t)e * (Ec * Mc)];
    ar += xr * wrv - xi * wiv;
    ai += xr * wiv + xi * wrv;
  }
  float* ob = S2 + ((size_t)(bn * Dc + h * Ec + o)) * 64 + m;
  ob[0]  = ar;
  ob[32] = ai;
}

// ---------------- LDS-staged double-buffered WMMA-f32 GEMM ------------------
// C = A(MxK) @ W(NxK)^T [+bias][+R].  Block = 128 threads (4 waves),
// block tile 64 rows x 64 cols, K consumed in KB=32 blocks.
// Per K-block: cooperative b128 global loads stage A/W tiles into LDS (next
// buffer) while WMMAs consume the current buffer from LDS (ds_load_b64, padded
// stride => conflict-free).  Global latency is hidden behind 128 WMMAs/block.
__global__ void gemm_k(const float* __restrict__ A, int lda,
                       const float* __restrict__ Wt, const float* __restrict__ bias,
                       const float* __restrict__ Rres, float* __restrict__ Cout,
                       int M, int N, int K) {
  __shared__ __align__(16) float as[2][64 * LDP];
  __shared__ __align__(16) float ws[2][64 * LDP];
  int tid = threadIdx.x;
  int wave = tid >> 5, lane = tid & 31;
  int lm = lane & 15, lh = lane >> 4;
  int row0 = blockIdx.x * 64;
  int col0 = blockIdx.y * 64;

  // cooperative-load task map: 512 float4 tasks per operand, 4 per thread
  int trow[4], tc[4];
  const float* agp[4];
  const float* wgp[4];
  #pragma unroll
  for (int i = 0; i < 4; ++i) {
    int task = tid + 128 * i;
    trow[i] = task >> 3;           // 0..63 (local row / local col)
    tc[i]   = (task & 7) * 4;      // float4 offset inside K-block
    int r = row0 + trow[i]; r = (r < M) ? r : (M - 1);
    agp[i] = A + (size_t)r * lda + tc[i];
    int n = col0 + trow[i]; n = (n < N) ? n : (N - 1);
    wgp[i] = Wt + (size_t)n * K + tc[i];
  }

  // stage K-block 0
  v4f abuf[4], wbuf[4];
  #pragma unroll
  for (int i = 0; i < 4; ++i) {
    abuf[i] = *(const v4f*)(agp[i]);
    wbuf[i] = *(const v4f*)(wgp[i]);
  }
  #pragma unroll
  for (int i = 0; i < 4; ++i) {
    float* da = &as[0][trow[i] * LDP + tc[i]];
    float* dw = &ws[0][trow[i] * LDP + tc[i]];
    da[0] = abuf[i][0]; da[1] = abuf[i][1]; da[2] = abuf[i][2]; da[3] = abuf[i][3];
    dw[0] = wbuf[i][0]; dw[1] = wbuf[i][1]; dw[2] = wbuf[i][2]; dw[3] = wbuf[i][3];
  }
  __syncthreads();

  v8f acc[4] = {};
  int nK = K / KB;
  for (int kb = 0; kb < nK; ++kb) {
    int cur = kb & 1, nxt = cur ^ 1;
    // issue global loads for the next K-block (drained only at the LDS store)
    if (kb + 1 < nK) {
      #pragma unroll
      for (int i = 0; i < 4; ++i) {
        abuf[i] = *(const v4f*)(agp[i] + (kb + 1) * KB);
        wbuf[i] = *(const v4f*)(wgp[i] + (kb + 1) * KB);
      }
    }
    if (kb + 2 < nK) {   // keep L2 warm two blocks ahead
      __builtin_prefetch(agp[0] + (kb + 2) * KB, 0, 1);
      __builtin_prefetch(wgp[0] + (kb + 2) * KB, 0, 1);
    }
    // compute current K-block entirely from LDS
    const float* asb = &as[cur][(wave * 16 + lm) * LDP + 2 * lh];
    #pragma unroll
    for (int ks = 0; ks < KB / 4; ++ks) {
      v2f a = *(const v2f*)(asb + 4 * ks);
      #pragma unroll
      for (int t = 0; t < 4; ++t) {
        const float* wsb = &ws[cur][(t * 16 + lm) * LDP + 2 * lh];
        v2f b = *(const v2f*)(wsb + 4 * ks);
        acc[t] = wmma4(a, b, acc[t]);
      }
    }
    // park the prefetched tiles into the other LDS buffer
    if (kb + 1 < nK) {
      #pragma unroll
      for (int i = 0; i < 4; ++i) {
        float* da = &as[nxt][trow[i] * LDP + tc[i]];
        float* dw = &ws[nxt][trow[i] * LDP + tc[i]];
        da[0] = abuf[i][0]; da[1] = abuf[i][1]; da[2] = abuf[i][2]; da[3] = abuf[i][3];
        dw[0] = wbuf[i][0]; dw[1] = wbuf[i][1]; dw[2] = wbuf[i][2]; dw[3] = wbuf[i][3];
      }
    }
    __syncthreads();
  }

  int rw0 = row0 + wave * 16;
  #pragma unroll
  for (int t = 0; t < 4; ++t) {
    int n = col0 + t * 16 + lm;
    if (n >= N) continue;
    float bs = bias ? bias[n] : 0.0f;
    #pragma unroll
    for (int i = 0; i < 8; ++i) {
      int r = rw0 + i + 8 * lh;
      if (r >= M) continue;
      size_t off = (size_t)r * N + n;
      float v = acc[t][i] + bs;
      if (Rres) v += Rres[off];
      Cout[off] = v;
    }
  }
}

// ---------------- fused FFN: out = X + gelu(X@c1w^T)@c2w^T ------------------
// 4 waves/block, each wave owns a 16-row x 128-col tile; DFF chunked by 16.
// The X operand (K=128) is loaded ONCE into 32 v2f registers and reused for
// all 32 DFF chunks; streamed c1w/c2w fragments are loaded in groups ahead of
// their WMMAs.
__global__ void ffn_k(const float* __restrict__ X, const float* __restrict__ c1w,
                      const float* __restrict__ c2w, float* __restrict__ out) {
  __shared__ __align__(16) float hb[4][256];
  int wave = threadIdx.x >> 5;
  int lane = threadIdx.x & 31;
  int lm = lane & 15, lh = lane >> 4;
  int row0 = blockIdx.x * 64 + wave * 16;
  const float* xrow = X + (size_t)(row0 + lm) * Dc + 2 * lh;
  float* hw = hb[wave];

  v2f xa[32];                     // resident A-fragments for the whole K=128
  #pragma unroll
  for (int ks = 0; ks < 32; ++ks) xa[ks] = *(const v2f*)(xrow + 4 * ks);

  v8f y[8] = {};
  for (int f0 = 0; f0 < DFFc; f0 += 16) {
    // h_chunk(16x16) = X_tile(16x128) @ c1w[f0:f0+16, :]^T
    v8f h = {};
    const float* w1p = c1w + (size_t)(f0 + lm) * Dc + 2 * lh;
    #pragma unroll
    for (int ks8 = 0; ks8 < 32; ks8 += 8) {
      v2f b[8];
      #pragma unroll
      for (int j = 0; j < 8; ++j) b[j] = *(const v2f*)(w1p + 4 * (ks8 + j));
      #pragma unroll
      for (int j = 0; j < 8; ++j) h = wmma4(xa[ks8 + j], b[j], h);
    }
    // gelu, bounce through LDS to reshape C-fragment -> A-fragment
    #pragma unroll
    for (int i = 0; i < 8; ++i)
      hw[(i + 8 * lh) * 16 + lm] = gelu_exact(h[i]);
    __syncthreads();
    v2f af[4];
    #pragma unroll
    for (int j = 0; j < 4; ++j)
      af[j] = *(const v2f*)(hw + lm * 16 + 4 * j + 2 * lh);
    // y_tile(16x128) += h_chunk(16x16) @ c2w[:, f0:f0+16]^T
    #pragma unroll
    for (int ct = 0; ct < 8; ++ct) {
      const float* w2p = c2w + (size_t)(ct * 16 + lm) * DFFc + f0 + 2 * lh;
      v2f b[4];
      #pragma unroll
      for (int j = 0; j < 4; ++j) b[j] = *(const v2f*)(w2p + 4 * j);
      #pragma unroll
      for (int j = 0; j < 4; ++j) y[ct] = wmma4(af[j], b[j], y[ct]);
    }
    __syncthreads();
  }
  #pragma unroll
  for (int ct = 0; ct < 8; ++ct) {
    #pragma unroll
    for (int i = 0; i < 8; ++i) {
      int r = row0 + i + 8 * lh;
      int c = ct * 16 + lm;
      size_t off = (size_t)r * Dc + c;
      out[off] = y[ct][i] + X[off];
    }
  }
}

// ---------------- layernorm stage 1: per-token LN over D --------------------
__global__ void ln1_k(const float* __restrict__ X, const float* __restrict__ g,
                      const float* __restrict__ be, float* __restrict__ xh) {
  __shared__ float red[Dc];
  int row = blockIdx.x;
  int tid = threadIdx.x;
  float v = X[(size_t)row * Dc + tid];
  red[tid] = v;
  __syncthreads();
  for (int s = Dc / 2; s > 0; s >>= 1) {
    if (tid < s) red[tid] += red[tid + s];
    __syncthreads();
  }
  float mu = red[0] * (1.0f / Dc);
  __syncthreads();
  float dv = v - mu;
  red[tid] = dv * dv;
  __syncthreads();
  for (int s = Dc / 2; s > 0; s >>= 1) {
    if (tid < s) red[tid] += red[tid + s];
    __syncthreads();
  }
  float var = red[0] * (1.0f / Dc);
  xh[(size_t)row * Dc + tid] = g[tid] * dv * rsqrtf(var + 1e-5f) + be[tid];
}

// ---------------- layernorm stage 2: subtract mean over L -------------------
__global__ void ln2_k(const float* __restrict__ xh, float* __restrict__ out) {
  int idx = blockIdx.x * blockDim.x + threadIdx.x;
  if (idx >= FR) return;
  int d = idx & (Dc - 1);
  int bn = idx >> 7;
  const float* base = xh + (size_t)bn * Lc * Dc + d;
  float* ob = out + (size_t)bn * Lc * Dc + d;
  float s = 0.0f;
  for (int l = 0; l < Lc; ++l) s += base[(size_t)l * Dc];
  float mu = s * (1.0f / Lc);
  for (int l = 0; l < Lc; ++l) ob[(size_t)l * Dc] = base[(size_t)l * Dc] - mu;
}

// ---------------- trend head: (BN,L) @ (PRED,L)^T + b -----------------------
__global__ void trend_head_k(const float* __restrict__ trend0, const float* __restrict__ tw,
                             const float* __restrict__ tb, float* __restrict__ tbuf) {
  int idx = blockIdx.x * blockDim.x + threadIdx.x;
  if (idx >= BNc * PREDc) return;
  int p = idx % PREDc;
  int bn = idx / PREDc;
  int b = bn / Nc;
  int n = bn - b * Nc;
  const float* tr = trend0 + (size_t)b * Lc * Nc + n;
  const float* w = tw + (size_t)p * Lc;
  float s = tb[p];
  for (int l = 0; l < Lc; ++l) s += tr[(size_t)l * Nc] * w[l];
  tbuf[idx] = s;
}

// ---------------- final combine: (B,PRED,N) = seas + trend ------------------
__global__ void combine_k(const float* __restrict__ sbuf, const float* __restrict__ tbuf,
                          float* __restrict__ out) {
  int idx = blockIdx.x * blockDim.x + threadIdx.x;
  if (idx >= Bc * PREDc * Nc) return;
  int n = idx % Nc;
  int r = idx / Nc;
  int p = r % PREDc;
  int b = r / PREDc;
  size_t src = ((size_t)(b * Nc + n)) * PREDc + p;
  out[idx] = sbuf[src] + tbuf[src];
}

// ============================ host driver ====================================
extern "C" void kernel_launch(void* const* d_in, const int* in_sizes, int n_in,
                              void* d_out, int out_size, void* d_ws, size_t ws_size,
                              hipStream_t stream) {
  const float* x_enc   = (const float*)d_in[0];
  const float* x_mark  = (const float*)d_in[1];
  const float* tok_w   = (const float*)d_in[4];
  const float* time_w  = (const float*)d_in[5];
  const float* qw      = (const float*)d_in[6];
  const float* qb      = (const float*)d_in[7];
  const float* ow      = (const float*)d_in[12];
  const float* ob      = (const float*)d_in[13];
  const float* f_wr    = (const float*)d_in[14];
  const float* f_wi    = (const float*)d_in[15];
  const float* c1w     = (const float*)d_in[16];
  const float* c2w     = (const float*)d_in[17];
  const float* gamma   = (const float*)d_in[18];
  const float* beta    = (const float*)d_in[19];
  const float* head_w  = (const float*)d_in[20];
  const float* head_b  = (const float*)d_in[21];
  const float* trend_w = (const float*)d_in[22];
  const float* trend_b = (const float*)d_in[23];
  float* outp = (float*)d_out;

  // workspace carve-up (floats)
  float* wsf = (float*)d_ws;
  size_t off = 0;
  const size_t ACT = (size_t)ROWS * Dc;   // 22,020,096
  float* X   = wsf + off; off += ACT;
  float* BA  = wsf + off; off += ACT;
  float* BB  = wsf + off; off += ACT;
  float* S1  = wsf + off; off += (size_t)FR * 64;
  float* S2  = wsf + off; off += (size_t)FR * 64;
  float* W1  = wsf + off; off += 64 * Lc;     // DFT basis (64 x 512)
  float* W3  = wsf + off; off += Lc * 64;     // iDFT basis (512 x 64)
  float* seas0  = wsf + off; off += (size_t)Bc * Lc * Nc;
  float* trend0 = wsf + off; off += (size_t)Bc * Lc * Nc;
  float* tbuf = wsf + off; off += (size_t)BNc * PREDc;
  float* sbuf = wsf + off; off += (size_t)BNc * PREDc;

  // 0) DFT / iDFT bases
  basis_k<<<(64 * Lc + Lc * 64 + 255) / 256, 256, 0, stream>>>(W1, W3);
  // 1) initial decomposition of x_enc (B,L,N)
  decomp_k<<<(Bc * Nc + 255) / 256, 256, 0, stream>>>(x_enc, seas0, trend0, Bc, Nc);
  // 2) trend head -> tbuf (BN,PRED)
  trend_head_k<<<(BNc * PREDc + 255) / 256, 256, 0, stream>>>(trend0, trend_w, trend_b, tbuf);
  // 3) embedding -> X (BN,L,D)
  embed_k<<<(BNc * Lc * Dc) / 256, 256, 0, stream>>>(seas0, x_mark, tok_w, time_w, X);

  for (int l = 0; l < NLc; ++l) {
    const float* qw_l = qw + (size_t)l * Dc * Dc;
    const float* qb_l = qb + (size_t)l * Dc;
    const float* ow_l = ow + (size_t)l * Dc * Dc;
    const float* ob_l = ob + (size_t)l * Dc;
    const float* wr_l = f_wr + (size_t)l * Hc * Ec * Ec * Mc;
    const float* wi_l = f_wi + (size_t)l * Hc * Ec * Ec * Mc;
    const float* c1_l = c1w + (size_t)l * DFFc * Dc;
    const float* c2_l = c2w + (size_t)l * Dc * DFFc;

    // q = X @ qw^T + qb        -> BA (BN,L,D)
    gemm_k<<<dim3(ROWS / 64, Dc / 64), 128, 0, stream>>>(X, Dc, qw_l, qb_l, nullptr, BA, ROWS, Dc, Dc);
    // transpose (BN,L,D) -> (BN,D,L)  (q -> per-(h,e) time series)
    tr021_k<<<dim3(Dc / 32, Lc / 32, BNc), dim3(32, 8), 0, stream>>>(BA, BB, Lc, Dc);
    // truncated rfft: S1 = BB(FR x L) @ W1^T  (cols = [Re|Im], 32 modes)
    gemm_k<<<dim3(FR / 64, 1), 128, 0, stream>>>(BB, Lc, W1, nullptr, nullptr, S1, FR, 64, Lc);
    // complex mixing over e (per head, per mode)
    mix_k<<<(BNc * Hc * Ec * Mc + 255) / 256, 256, 0, stream>>>(S1, wr_l, wi_l, S2);
    // irfft of 32-mode spectrum: BB = S2(FR x 64) @ W3^T  -> (bn,h,o,l) flat,
    // which the reference's raw reshape reads directly as (BN,L,D).
    gemm_k<<<dim3(FR / 64, Lc / 64), 128, 0, stream>>>(S2, 64, W3, nullptr, nullptr, BB, FR, Lc, 64);
    // x = X + att @ ow^T + ob  -> BA
    gemm_k<<<dim3(ROWS / 64, Dc / 64), 128, 0, stream>>>(BB, Dc, ow_l, ob_l, X, BA, ROWS, Dc, Dc);
    // x = seasonal(x)
    decomp_k<<<(BNc * Dc + 255) / 256, 256, 0, stream>>>(BA, X, nullptr, BNc, Dc);
    // BA = x + gelu(x@c1w^T)@c2w^T  (fused, DFF never materialized)
    ffn_k<<<ROWS / 64, 128, 0, stream>>>(X, c1_l, c2_l, BA);
    // x = seasonal(x + y)
    decomp_k<<<(BNc * Dc + 255) / 256, 256, 0, stream>>>(BA, X, nullptr, BNc, Dc);
  }

  // final layernorm (per-token LN, then subtract per-(bn,d) mean over L)
  ln1_k<<<ROWS, Dc, 0, stream>>>(X, gamma, beta, BB);
  ln2_k<<<(FR + 255) / 256, 256, 0, stream>>>(BB, BA);
  // seasonal head: (BN, L*D) @ head_w^T + head_b -> sbuf (BN,PRED)
  gemm_k<<<dim3((BNc + 63) / 64, (PREDc + 63) / 64), 128, 0, stream>>>(
      BA, Lc * Dc, head_w, head_b, nullptr, sbuf, BNc, PREDc, Lc * Dc);
  // out (B,PRED,N) = seasonal + trend
  combine_k<<<(Bc * PREDc * Nc + 255) / 256, 256, 0, stream>>>(sbuf, tbuf, outp);
  (void)in_sizes; (void)n_in; (void)out_size; (void)ws_size;
}